// SpectralConv1d_10196252360929
// MI455X (gfx1250) — compile-verified
//
#include <hip/hip_runtime.h>

// Problem constants (from reference setup_inputs): B=32, Ci=Co=128, N=4096, M=1024
constexpr int B  = 32;
constexpr int CI = 128;
constexpr int CO = 128;
constexpr int N  = 4096;   // 2^12
constexpr int LOGN = 12;
constexpr int M  = 1024;   // modes kept

typedef float v2f __attribute__((ext_vector_type(2)));
typedef float v8f __attribute__((ext_vector_type(8)));
typedef unsigned int u32x4 __attribute__((ext_vector_type(4)));
typedef int i32x4 __attribute__((ext_vector_type(4)));
typedef int i32x8 __attribute__((ext_vector_type(8)));

#if defined(__has_builtin)
#if __has_builtin(__builtin_amdgcn_tensor_load_to_lds) && \
    __has_builtin(__builtin_amdgcn_s_wait_tensorcnt)
#define USE_TDM 1
#endif
#endif

// ---------------------------------------------------------------------------
// Kernel 0: transpose weights (Ci, Co, M) -> wt (M, Ci, Co) so the per-mode
// 128x128 W slab is a contiguous 64KB coalesced read for the GEMM kernel.
// ---------------------------------------------------------------------------
__global__ __launch_bounds__(256) void k_transpose_w(const float* __restrict__ w,
                                                     float* __restrict__ wt) {
    __shared__ float tile[32][33];                 // +1 pad kills bank conflicts
    const int i  = blockIdx.z;                     // Ci index
    const int o0 = blockIdx.y * 32;                // Co tile
    const int k0 = blockIdx.x * 32;                // M tile
    const int tx = threadIdx.x & 31;
    const int ty = threadIdx.x >> 5;               // 8 rows of 32
    #pragma unroll
    for (int r = ty; r < 32; r += 8)               // coalesced along k
        tile[r][tx] = w[((size_t)i * CO + (o0 + r)) * M + (k0 + tx)];
    __syncthreads();
    #pragma unroll
    for (int r = ty; r < 32; r += 8)               // coalesced along o
        wt[((size_t)(k0 + r) * CI + i) * CO + (o0 + tx)] = tile[tx][r];
}

// ---------------------------------------------------------------------------
// Kernel 1: forward FHT of each x row (4096 rows of length 4096) in LDS.
// Writes only the first M modes, transposed: xt[k][b*Ci + i].
// ---------------------------------------------------------------------------
__global__ __launch_bounds__(256) void k_fht_x(const float* __restrict__ x,
                                               float* __restrict__ xt) {
    __shared__ float y[N];                          // 16 KB
    const int r = blockIdx.x;                       // r = b*Ci + i
    const float* xr = x + (size_t)r * N;
    // bit-reversal load: coalesced global read, scattered LDS write
    for (int j = threadIdx.x; j < N; j += 256) {
        const unsigned rev = __brev((unsigned)j) >> (32 - LOGN);
        y[rev] = xr[j];
    }
    __syncthreads();
    #pragma unroll 1
    for (int s = 1; s <= LOGN; ++s) {
        const int m    = 1 << s;
        const int half = m >> 1;
        const float w0 = 6.28318530717958647692f / (float)m;
        for (int t = threadIdx.x; t < N / 2; t += 256) {
            const int blk = t >> (s - 1);
            const int n   = t & (half - 1);
            const int p0  = blk * m + n;
            const int p1  = p0 + half;
            const float f   = w0 * (float)n;
            const float cas = __cosf(f) + __sinf(f);
            const float E = y[p0], O = y[p1];
            y[p0] = E + cas * O;
            y[p1] = E - cas * O;
        }
        __syncthreads();
    }
    for (int k = threadIdx.x; k < M; k += 256)
        xt[(size_t)k * (B * CI) + r] = y[k];
}

// ---------------------------------------------------------------------------
// Kernel 2: per-mode GEMM with f32 WMMA.
//   D(32x128) = 0.5 * ( (A+An)(32x128) * Wk(128x128) + (A-An) * Wkn )
// One block per mode k; 16 wave32 -> 2x8 grid of 16x16 WMMA tiles; K in
// steps of 4 via V_WMMA_F32_16X16X4_F32. LDS (dynamic, 160KB):
//   Wk[128][128] @0 | Wkn[128][128] @64KB | Ap[32][128] | Am[32][128]
// W slabs are DMA'd by the Tensor Data Mover (TENSORcnt) while the A slabs
// are staged with normal loads; result scattered into d_out[b][o*N + k].
// ---------------------------------------------------------------------------
__global__ __launch_bounds__(512) void k_mode_gemm(const float* __restrict__ xt,
                                                   const float* __restrict__ wt,
                                                   float* __restrict__ out) {
    extern __shared__ float lds[];
    float* wk  = lds;                      // 128*128 @ LDS byte offset 0
    float* wkn = lds + CI * CO;            // 128*128 @ LDS byte offset 65536
    float* ap  = lds + 2 * CI * CO;        // 32*128
    float* am  = ap + B * CI;              // 32*128

    const int k  = blockIdx.x;
    const int kn = (M - k) & (M - 1);      // (M-k) % M, M power of two

#ifdef USE_TDM
    // Tensor Data Mover: DMA both 64KB W slabs straight into LDS (no VGPRs).
    // D# group1: data_size=4B (bits17:16=2); tensor_dim0=128 (bits79:48),
    // tensor_dim1=128 (bits111:80), tile_dim0=128 (bits127:112),
    // tile_dim1=128 (bits143:128), tensor_dim0_stride=128 (bits207:160).
    if (threadIdx.x < 32) {                // one wave issues both descriptors
        const i32x8 g1 = {0x20000, 128 << 16, 128 << 16, 128 << 16,
                          128, 128, 0, 0};
        const i32x4 z4 = {0, 0, 0, 0};
        const i32x8 z8 = {0, 0, 0, 0, 0, 0, 0, 0};
        const unsigned long long gak =
            (unsigned long long)(const void*)(wt + (size_t)k  * CI * CO);
        const unsigned long long gan =
            (unsigned long long)(const void*)(wt + (size_t)kn * CI * CO);
        // group0: {count=1, lds_addr, global_addr[31:0], ga[56:32] | type=2<<30}
        // (dynamic LDS base is offset 0: no static LDS in this kernel)
        const u32x4 g0k = {1u, 0u, (unsigned)gak,
                           (unsigned)(gak >> 32) | (2u << 30)};
        const u32x4 g0n = {1u, (unsigned)(CI * CO * sizeof(float)), (unsigned)gan,
                           (unsigned)(gan >> 32) | (2u << 30)};
        __builtin_amdgcn_tensor_load_to_lds(g0k, g1, z4, z4, z8, 0);
        __builtin_amdgcn_tensor_load_to_lds(g0n, g1, z4, z4, z8, 0);
    }
#else
    // Fallback: stage W slabs with coalesced float4 loads
    {
        const float4* gwk  = (const float4*)(wt + (size_t)k  * CI * CO);
        const float4* gwkn = (const float4*)(wt + (size_t)kn * CI * CO);
        for (int idx = threadIdx.x; idx < (CI * CO) / 4; idx += 512) {
            ((float4*)wk)[idx]  = gwk[idx];
            ((float4*)wkn)[idx] = gwkn[idx];
        }
    }
#endif

    // stage A slabs (16KB) while the TDM runs, forming plus/minus combos
    const float* gxk  = xt + (size_t)k  * (B * CI);
    const float* gxkn = xt + (size_t)kn * (B * CI);
    for (int idx = threadIdx.x; idx < B * CI; idx += 512) {
        const float a  = gxk[idx];
        const float an = gxkn[idx];
        ap[idx] = a + an;
        am[idx] = a - an;
    }

#ifdef USE_TDM
    if (threadIdx.x < 32)
        __builtin_amdgcn_s_wait_tensorcnt(0);  // TDM done before barrier
#endif
    __syncthreads();

    const int wave = threadIdx.x >> 5;
    const int lane = threadIdx.x & 31;
    const int mt   = wave & 1;             // 2 row-tiles of 16 (b)
    const int nt   = wave >> 1;            // 8 col-tiles of 16 (o)
    const int row  = mt * 16 + (lane & 15);
    const int col  = nt * 16 + (lane & 15);
    const int koff = (lane < 16) ? 0 : 2;  // f32 16x4 A fragment: half-wave K split

    v8f acc = {};
    #pragma unroll 8
    for (int kk = 0; kk < CI; kk += 4) {
        // A fragments: two consecutive K values per lane -> 8B LDS load
        v2f afp = *(const v2f*)(ap + row * CI + kk + koff);
        v2f afm = *(const v2f*)(am + row * CI + kk + koff);
        // B fragments: lane holds column `col`, rows kk+koff, kk+koff+1
        v2f bfk, bfn;
        bfk.x = wk [(kk + koff)     * CO + col];
        bfk.y = wk [(kk + koff + 1) * CO + col];
        bfn.x = wkn[(kk + koff)     * CO + col];
        bfn.y = wkn[(kk + koff + 1) * CO + col];
        acc = __builtin_amdgcn_wmma_f32_16x16x4_f32(false, afp, false, bfk,
                                                    (short)0, acc, false, false);
        acc = __builtin_amdgcn_wmma_f32_16x16x4_f32(false, afm, false, bfn,
                                                    (short)0, acc, false, false);
    }

    // D layout: VGPR j -> row (mt*16 + j + 8*(lane>=16)), col = nt*16 + (lane&15)
    const int rbase = mt * 16 + ((lane < 16) ? 0 : 8);
    #pragma unroll
    for (int j = 0; j < 8; ++j) {
        const int b = rbase + j;
        out[((size_t)(b * CO + col)) * N + k] = 0.5f * acc[j];
    }
}

// ---------------------------------------------------------------------------
// Kernel 3: inverse FHT (same transform) on each output row, in place.
// Reads only the first M entries of the row, zero-pads in LDS, scales by 1/N.
// ---------------------------------------------------------------------------
__global__ __launch_bounds__(256) void k_fht_out(float* __restrict__ out) {
    __shared__ float y[N];
    const int r = blockIdx.x;              // r = b*Co + o
    float* orow = out + (size_t)r * N;
    for (int j = threadIdx.x; j < N; j += 256) {
        const unsigned rev = __brev((unsigned)j) >> (32 - LOGN);
        y[rev] = (j < M) ? orow[j] : 0.0f;
    }
    __syncthreads();
    #pragma unroll 1
    for (int s = 1; s <= LOGN; ++s) {
        const int m    = 1 << s;
        const int half = m >> 1;
        const float w0 = 6.28318530717958647692f / (float)m;
        for (int t = threadIdx.x; t < N / 2; t += 256) {
            const int blk = t >> (s - 1);
            const int n   = t & (half - 1);
            const int p0  = blk * m + n;
            const int p1  = p0 + half;
            const float f   = w0 * (float)n;
            const float cas = __cosf(f) + __sinf(f);
            const float E = y[p0], O = y[p1];
            y[p0] = E + cas * O;
            y[p1] = E - cas * O;
        }
        __syncthreads();
    }
    const float invn = 1.0f / (float)N;
    for (int j = threadIdx.x; j < N; j += 256)
        orow[j] = y[j] * invn;
}

// ---------------------------------------------------------------------------
extern "C" void kernel_launch(void* const* d_in, const int* in_sizes, int n_in,
                              void* d_out, int out_size, void* d_ws, size_t ws_size,
                              hipStream_t stream) {
    (void)in_sizes; (void)n_in; (void)out_size; (void)ws_size;
    const float* x = (const float*)d_in[0];       // (B, Ci, N)
    const float* w = (const float*)d_in[1];       // (Ci, Co, M)
    float* out = (float*)d_out;                   // (B, Co, N)

    float* wt = (float*)d_ws;                     // (M, Ci, Co)  64 MB
    float* xt = (float*)d_ws + (size_t)M * CI * CO; // (M, B*Ci)  16 MB

    k_transpose_w<<<dim3(M / 32, CO / 32, CI), 256, 0, stream>>>(w, wt);
    k_fht_x<<<B * CI, 256, 0, stream>>>(x, xt);

    const size_t lds_bytes = (size_t)(2 * CI * CO + 2 * B * CI) * sizeof(float); // 160 KB
    k_mode_gemm<<<M, 512, lds_bytes, stream>>>(xt, wt, out);

    k_fht_out<<<B * CO, 256, 0, stream>>>(out);
}